// DiffKS_53669911331128
// MI455X (gfx1250) — compile-verified
//
#include <hip/hip_runtime.h>
#include <hip/hip_bf16.h>
#include <math.h>

#define N_FRAMES 250
#define NSEG 249
#define N_SAMPLES_C 131072
#define RING 1024
#define THREADS_SCAN 128
#define SSTAGE 1024                 // samples staged per window (n % SSTAGE == 0)

// Async global->LDS path (gfx1250). Evaluated false on the host pass; helpers
// degrade to plain copies there (host never codegens device bodies anyway).
#if defined(__HIP_DEVICE_COMPILE__) && \
    __has_builtin(__builtin_amdgcn_global_load_async_to_lds_b32) && \
    __has_builtin(__builtin_amdgcn_s_wait_asynccnt)
#define KS_ASYNC 1
#else
#define KS_ASYNC 0
#endif

// Prefer one B128 async op per 16B; fall back to 4x B32.
#if KS_ASYNC && __has_builtin(__builtin_amdgcn_global_load_async_to_lds_b128)
#define KS_OPS16 1
#elif KS_ASYNC
#define KS_OPS16 4
#else
#define KS_OPS16 0
#endif

// sweeps * 5 arrays * ops-per-16B = async ops issued per thread per window
#define KS_SWEEPS (SSTAGE / (THREADS_SCAN * 4))
#define KS_OPS_PER_WIN (KS_SWEEPS * 5 * KS_OPS16)

typedef int ks_v4i __attribute__((vector_size(16)));   // b128 operand type

// copy 16 bytes (4 dwords) global -> LDS, asynchronously when available
__device__ __forceinline__ void ks_async_cp16(const void* g, void* l) {
#if KS_ASYNC
#if KS_OPS16 == 1
  __attribute__((address_space(1))) ks_v4i* gp =
      (__attribute__((address_space(1))) ks_v4i*)const_cast<void*>(g);
  __attribute__((address_space(3))) ks_v4i* lp =
      (__attribute__((address_space(3))) ks_v4i*)l;
  __builtin_amdgcn_global_load_async_to_lds_b128(gp, lp, 0, 0);
#else
  __attribute__((address_space(1))) int* gp =
      (__attribute__((address_space(1))) int*)const_cast<void*>(g);
  __attribute__((address_space(3))) int* lp =
      (__attribute__((address_space(3))) int*)l;
  __builtin_amdgcn_global_load_async_to_lds_b32(gp + 0, lp + 0, 0, 0);
  __builtin_amdgcn_global_load_async_to_lds_b32(gp + 1, lp + 1, 0, 0);
  __builtin_amdgcn_global_load_async_to_lds_b32(gp + 2, lp + 2, 0, 0);
  __builtin_amdgcn_global_load_async_to_lds_b32(gp + 3, lp + 3, 0, 0);
#endif
#else
  const int* gs = (const int*)g;
  int* ld = (int*)l;
  ld[0] = gs[0]; ld[1] = gs[1]; ld[2] = gs[2]; ld[3] = gs[3];
#endif
}

// wait until previously-staged window is complete (in-order ASYNCcnt)
__device__ __forceinline__ void ks_wait_prev(bool issued_next) {
#if KS_ASYNC
  if (issued_next) __builtin_amdgcn_s_wait_asynccnt(KS_OPS_PER_WIN);
  else             __builtin_amdgcn_s_wait_asynccnt(0);
#endif
}

// Knot positions must be identical everywhere: t_i = i/249 (f32), endpoints exact.
__device__ __forceinline__ float knotf(int i) {
  if (i <= 0) return 0.0f;
  if (i >= NSEG) return 1.0f;
  return (float)((double)i / (double)NSEG);
}

// ---------------------------------------------------------------------------
// Kernel A: sigmoid-normalize coeff knots + natural cubic spline fit (Thomas
// algorithm on the SPD tridiagonal system, 3 channels). Writes per-interval
// coeffs a, b, two_c, three_d to ws: sc[(ch*4+q)*256 + i].
// ---------------------------------------------------------------------------
__global__ __launch_bounds__(256) void ks_spline_fit(
    const float* __restrict__ delay_frames,
    const float* __restrict__ raw_coeff,
    float* __restrict__ sc,
    int* __restrict__ minz) {
  __shared__ float s_hinv[NSEG];
  __shared__ float s_x[3][N_FRAMES];
  __shared__ float s_k[3][N_FRAMES];
  __shared__ float s_cp[3][N_FRAMES];
  __shared__ float s_dp[3][N_FRAMES];
  const int tid = threadIdx.x;

  for (int i = tid; i < N_FRAMES; i += blockDim.x) {
    s_x[0][i] = delay_frames[i];
    float r0 = raw_coeff[2 * i + 0];
    float r1 = raw_coeff[2 * i + 1];
    float s0 = 1.0f / (1.0f + expf(-r0));
    float s1 = 1.0f / (1.0f + expf(-r1));
    float inv = 1.0f / (s0 + s1);
    s_x[1][i] = s0 * inv;
    s_x[2][i] = s1 * inv;
  }
  for (int i = tid; i < NSEG; i += blockDim.x)
    s_hinv[i] = 1.0f / (knotf(i + 1) - knotf(i));
  __syncthreads();

  if (tid < 3) {           // one lane per channel; 250-step serial Thomas solve
    const int ch = tid;
    float hi0 = s_hinv[0];
    float rp_prev = 3.0f * (s_x[ch][1] - s_x[ch][0]) * hi0 * hi0;
    float d0 = 2.0f * hi0;
    s_cp[ch][0] = hi0 / d0;
    s_dp[ch][0] = rp_prev / d0;
    for (int i = 1; i < N_FRAMES; ++i) {
      float lower = s_hinv[i - 1];
      float diag, rhs, upper;
      if (i < N_FRAMES - 1) {
        float hi = s_hinv[i];
        float rp = 3.0f * (s_x[ch][i + 1] - s_x[ch][i]) * hi * hi;
        diag = 2.0f * (lower + hi);
        rhs = rp_prev + rp;
        upper = hi;
        rp_prev = rp;
      } else {
        diag = 2.0f * lower;
        rhs = rp_prev;
        upper = 0.0f;
      }
      float m = 1.0f / (diag - lower * s_cp[ch][i - 1]);
      s_cp[ch][i] = upper * m;
      s_dp[ch][i] = (rhs - lower * s_dp[ch][i - 1]) * m;
    }
    float kk = s_dp[ch][N_FRAMES - 1];
    s_k[ch][N_FRAMES - 1] = kk;
    for (int i = N_FRAMES - 2; i >= 0; --i) {
      kk = s_dp[ch][i] - s_cp[ch][i] * kk;
      s_k[ch][i] = kk;
    }
  }
  __syncthreads();

  for (int i = tid; i < NSEG; i += blockDim.x) {
    float hc = s_hinv[i];
#pragma unroll
    for (int ch = 0; ch < 3; ++ch) {
      float x0 = s_x[ch][i], x1 = s_x[ch][i + 1];
      float k0 = s_k[ch][i], k1 = s_k[ch][i + 1];
      float dx3 = 3.0f * (x1 - x0);
      sc[(ch * 4 + 0) * 256 + i] = x0;                                              // a
      sc[(ch * 4 + 1) * 256 + i] = k0;                                              // b
      sc[(ch * 4 + 2) * 256 + i] = (2.0f * dx3 * hc - 4.0f * k0 - 2.0f * k1) * hc;  // two_c
      sc[(ch * 4 + 3) * 256 + i] = (-2.0f * dx3 * hc + 3.0f * (k0 + k1)) * hc * hc; // three_d
    }
  }
  if (tid == 0) *minz = 0x7fffffff;
}

// ---------------------------------------------------------------------------
// Kernel B: evaluate splines at all n sample points, derive z/g1/g2/g3/x,
// and block-reduce min(z) (spline overshoot can push z below 100, so the
// safe parallel chunk width for the scan must be derived from the data).
// ---------------------------------------------------------------------------
__global__ __launch_bounds__(256) void ks_eval(
    const float* __restrict__ exc, int exc_n,
    const float* __restrict__ sc,
    int* __restrict__ zA, float* __restrict__ g1A, float* __restrict__ g2A,
    float* __restrict__ g3A, float* __restrict__ xA,
    int* __restrict__ minz, int n) {
  __shared__ int red[256];
  const int tid = threadIdx.x;
  const int j = blockIdx.x * blockDim.x + tid;
  int zloc = 0x7fffffff;
  if (j < n) {
    float tout = (float)((double)j / (double)(n - 1));
    // emulate searchsorted(t_in, tout, 'left') - 1, clipped to [0, 248]
    int idx = (int)floorf(tout * (float)NSEG);
    idx = idx < 0 ? 0 : (idx > NSEG - 1 ? NSEG - 1 : idx);
    while (idx < NSEG - 1 && knotf(idx + 1) < tout) ++idx;
    while (idx > 0 && knotf(idx) >= tout) --idx;
    float f = tout - knotf(idx);
    float vals[3];
#pragma unroll
    for (int ch = 0; ch < 3; ++ch) {
      float a  = sc[(ch * 4 + 0) * 256 + idx];
      float b  = sc[(ch * 4 + 1) * 256 + idx];
      float c2 = sc[(ch * 4 + 2) * 256 + idx];
      float d3 = sc[(ch * 4 + 3) * 256 + idx];
      float inner = b + (0.5f * c2 + d3 * (f / 3.0f)) * f;
      vals[ch] = a + inner * f;
    }
    float delay = vals[0];
    float zf = floorf(delay);
    int z = (int)zf;
    float alfa = delay - zf;
    float b1 = vals[1], b2 = vals[2];
    zA[j]  = z;
    g1A[j] = b1 * (1.0f - alfa);
    g2A[j] = b1 * alfa + b2 * (1.0f - alfa);
    g3A[j] = b2 * alfa;
    xA[j]  = (j < exc_n) ? exc[j] : 0.0f;
    zloc = z;
  }
  red[tid] = zloc;
  __syncthreads();
  for (int s = 128; s > 0; s >>= 1) {
    if (tid < s) { int o = red[tid + s]; if (o < red[tid]) red[tid] = o; }
    __syncthreads();
  }
  if (tid == 0) atomicMin(minz, red[0]);
}

// ---------------------------------------------------------------------------
// Kernel C: the sequential recurrence. Min feedback lag = min(z)+1, so chunks
// of up to that many samples are fully parallel (1 workgroup, 4 wave32).
// y-history lives in a 1024-entry LDS ring. Coefficients are staged window-
// by-window into an LDS ping-pong buffer with GLOBAL_LOAD_ASYNC_TO_LDS_B128:
// the next window's DMA is issued before computing the current one, then
// s_wait_asynccnt(KS_OPS_PER_WIN) + barrier guarantees (in-order ASYNCcnt)
// that the current window is resident while the next streams in. The serial
// critical path per chunk is LDS latency + one barrier; no global-load
// latency ever sits on the dependency chain, and peak outstanding ASYNCcnt
// (2*KS_OPS_PER_WIN = 20 with b128) stays well under the 6-bit counter cap.
// ---------------------------------------------------------------------------
__global__ __launch_bounds__(THREADS_SCAN, 1) void ks_scan(
    const int* __restrict__ zA, const float* __restrict__ g1A,
    const float* __restrict__ g2A, const float* __restrict__ g3A,
    const float* __restrict__ xA, const int* __restrict__ minz,
    float* __restrict__ y, int n) {
  __shared__ float ring[RING];
  __shared__ alignas(16) int   sZ [2][SSTAGE];
  __shared__ alignas(16) float sG1[2][SSTAGE];
  __shared__ alignas(16) float sG2[2][SSTAGE];
  __shared__ alignas(16) float sG3[2][SSTAGE];
  __shared__ alignas(16) float sX [2][SSTAGE];
  const int tid = threadIdx.x;

  for (int i = tid; i < RING; i += THREADS_SCAN) ring[i] = 0.0f;

  int chunk = *minz + 1;
  if (chunk > THREADS_SCAN) chunk = THREADS_SCAN;
  if (chunk < 1) chunk = 1;

  // Stage one window: every thread issues exactly KS_OPS_PER_WIN async ops
  // (OOB global indices clamped so the op count stays fixed; those LDS slots
  // map to t >= n and are never read).
  auto stage = [&](int b, int base) {
#pragma unroll
    for (int s = 0; s < KS_SWEEPS; ++s) {
      int li = (tid + s * THREADS_SCAN) * 4;
      int gi = base + li;
      int gs = (gi + 4 <= n) ? gi : (n - 4);
      ks_async_cp16(&zA[gs],  &sZ [b][li]);
      ks_async_cp16(&g1A[gs], &sG1[b][li]);
      ks_async_cp16(&g2A[gs], &sG2[b][li]);
      ks_async_cp16(&g3A[gs], &sG3[b][li]);
      ks_async_cp16(&xA[gs],  &sX [b][li]);
    }
  };

  stage(0, 0);
  int buf = 0;

  for (int w0 = 0; w0 < n; w0 += SSTAGE) {
    bool has_next = (w0 + SSTAGE) < n;
    if (has_next) {
      stage(buf ^ 1, w0 + SSTAGE);
      // pull the window after next toward this WGP's caches
      int p = w0 + 2 * SSTAGE + tid * 8;
      if (p < n) {
        __builtin_prefetch(&zA[p],  0, 3);
        __builtin_prefetch(&g1A[p], 0, 3);
        __builtin_prefetch(&g2A[p], 0, 3);
        __builtin_prefetch(&g3A[p], 0, 3);
        __builtin_prefetch(&xA[p],  0, 3);
      }
    }
    ks_wait_prev(has_next);   // current window's async ops are done (in-order)
    __syncthreads();          // ...across all 4 waves (and ring init on iter 0)

    int wend = (w0 + SSTAGE < n) ? (w0 + SSTAGE) : n;
    for (int c = w0; c < wend; ) {
      int cur = chunk;
      if (c + cur > wend) cur = wend - c;   // truncated chunk is still safe
      int t = c + tid;
      if (tid < cur) {
        int li = t - w0;
        int zt = sZ[buf][li];
        float a1 = sG1[buf][li];
        float a2 = sG2[buf][li];
        float a3 = sG3[buf][li];
        float xt = sX[buf][li];
        int basei = t - zt;
        float y1 = ring[(basei - 1) & (RING - 1)];
        float y2 = ring[(basei - 2) & (RING - 1)];
        float y3 = ring[(basei - 3) & (RING - 1)];
        float yv = xt + a1 * y1;   // match reference left-to-right association
        yv += a2 * y2;
        yv += a3 * y3;
        ring[t & (RING - 1)] = yv; // write slots disjoint from all read slots
        y[t] = yv;
      }
      __syncthreads();
      c += cur;
    }
    buf ^= 1;
  }
}

// ---------------------------------------------------------------------------
extern "C" void kernel_launch(void* const* d_in, const int* in_sizes, int n_in,
                              void* d_out, int out_size, void* d_ws,
                              size_t ws_size, hipStream_t stream) {
  (void)n_in; (void)out_size; (void)ws_size;
  const float* delay = (const float*)d_in[0];   // (250,) f32
  const float* raw   = (const float*)d_in[1];   // (250,2) f32
  const float* exc   = (const float*)d_in[2];   // (2048,) f32
  const int n = N_SAMPLES_C;                    // reference N_SAMPLES constant
  const int exc_n = in_sizes[2];

  // workspace layout (floats): sc[3072] | minz[1]@3072 | z[n]@3328 | g1 g2 g3 x
  // (all array bases are 16B-aligned: 3328 and n are multiples of 4)
  float* wsF = (float*)d_ws;
  float* sc   = wsF;
  int*   minz = (int*)(wsF + 3072);
  int*   zA   = (int*)(wsF + 3328);
  float* g1A  = wsF + 3328 + n;
  float* g2A  = g1A + n;
  float* g3A  = g2A + n;
  float* xA   = g3A + n;

  ks_spline_fit<<<1, 256, 0, stream>>>(delay, raw, sc, minz);
  ks_eval<<<(n + 255) / 256, 256, 0, stream>>>(exc, exc_n, sc, zA, g1A, g2A,
                                               g3A, xA, minz, n);
  ks_scan<<<1, THREADS_SCAN, 0, stream>>>(zA, g1A, g2A, g3A, xA, minz,
                                          (float*)d_out, n);
}